// Spatial_Graph_Convolution_33277406609639
// MI455X (gfx1250) — compile-verified
//
#include <hip/hip_runtime.h>
#include <hip/hip_bf16.h>
#include <math.h>

typedef __attribute__((ext_vector_type(2))) float v2f;
typedef __attribute__((ext_vector_type(8))) float v8f;

#define BP      8192
#define NN      32
#define DD      256
#define KK      8
#define OO      32
#define WAVES   8
#define EPSF    1e-14f
#define TWO_PI_F 6.283185307179586f

// kernel_W [K][O][D] -> WT [K][D][O] so phase-2 B-operand loads are contiguous.
__global__ void wt_transpose_kernel(const float* __restrict__ W, float* __restrict__ WT) {
  int idx = blockIdx.x * 256 + threadIdx.x;           // 8*32*256 = 65536
  int k = idx >> 13;
  int o = (idx >> 8) & (OO - 1);
  int d = idx & (DD - 1);
  WT[k * (DD * OO) + d * OO + o] = W[idx];
}

__global__ __launch_bounds__(256) void sgc_fused_kernel(
    const float* __restrict__ in_feat,   // [BP][N][D]
    const float* __restrict__ polar,     // [BP][N][2]
    const float* __restrict__ mL,        // [K]
    const float* __restrict__ sdL,       // [K]
    const float* __restrict__ mA,        // [K]
    const float* __restrict__ WT,        // [K][D][O]
    float* __restrict__ out)             // [BP][K*O]
{
  // 64KB LDS: phase-2 layout is agg[k][d][b] (16384 floats).
  // gk transpose buffers are overlaid in the first 2048 floats; they are dead
  // for every wave before the barrier that precedes the agg stores.
  __shared__ float smem[KK * DD * WAVES];

  const int lane = threadIdx.x & 31;
  const int w    = threadIdx.x >> 5;     // wave id == b-row in block == k in phase 2
  const int half = lane >> 4;            // 0 / 1  (lanes 0-15 vs 16-31)
  const int l15  = lane & 15;
  const int b    = blockIdx.x * WAVES + w;

  // ---------------- Gaussian kernel weights (lane == neighbor n) ------------
  const float2 pc = ((const float2*)polar)[(size_t)b * NN + lane];
  float wgt[KK];
  float s = 0.f;
  #pragma unroll
  for (int k = 0; k < KK; ++k) {
    float ml = mL[k], sl = sdL[k], ma = mA[k];
    float dl = pc.x - ml;
    float dw = __expf(-0.5f * dl * dl / (EPSF + sl * sl));
    float da = fabsf(pc.y - ma);
    da = fminf(da, TWO_PI_F - da);
    float sa = EPSF + sl;                // reference (faithfully) uses (eps+sd)^2
    float aw = __expf(-0.5f * da * da / (sa * sa));
    float v = dw * aw;
    v = (v == v) ? v : 0.f;              // NaN guard
    wgt[k] = v;
    s += v;
  }
  const float inv = 1.f / s;
  float* gkw = smem + w * (NN * KK);     // per-wave 256-float transpose buffer
  #pragma unroll
  for (int k = 0; k < KK; ++k) gkw[lane * KK + k] = wgt[k] * inv;
  __syncthreads();

  // ---------------- Phase 1: agg = gk^T x feat via V_WMMA_F32_16X16X4_F32 ---
  // D tile t holds agg[k=0..7 (+8..15 junk)][d = 16t .. 16t+15]
  v8f acc[16];
  const v8f vzero = {0.f, 0.f, 0.f, 0.f, 0.f, 0.f, 0.f, 0.f};
  #pragma unroll
  for (int t = 0; t < 16; ++t) acc[t] = vzero;

  const float* fb = in_feat + (size_t)b * (NN * DD);
  for (int c = 0; c < 8; ++c) {          // n-chunks of 4
    const int r0 = 4 * c + 2 * half;     // A vgpr0 rows: kk=0 / kk=2
    v2f A;
    A.x = gkw[(r0 + 0) * KK + (l15 & 7)];   // A[m][kk]   (rows m>=8: don't-care)
    A.y = gkw[(r0 + 1) * KK + (l15 & 7)];
    const float* rp = fb + r0 * DD + l15;   // B[kk][nn] = feat[4c+kk][16t+nn]
    #pragma unroll
    for (int t = 0; t < 16; ++t) {
      v2f Bv;
      Bv.x = rp[t * 16];
      Bv.y = rp[DD + t * 16];
      acc[t] = __builtin_amdgcn_wmma_f32_16x16x4_f32(
          false, A, false, Bv, (short)0, acc[t], false, false);
    }
  }
  __syncthreads();   // all waves done reading gk overlay -> smem reusable

  // Store agg to LDS as [k][d][b]  (conflict-friendly for phase-2 A reads)
  if (lane < 16) {
    #pragma unroll
    for (int t = 0; t < 16; ++t)
      #pragma unroll
      for (int j = 0; j < KK; ++j)
        smem[j * (DD * WAVES) + (t * 16 + lane) * WAVES + w] = acc[t][j];
  }
  __syncthreads();

  // ---------------- Phase 2: out tile [8 b x 32 o] for kernel k = w ---------
  const float* wtk = WT + w * (DD * OO);
  v8f accO0 = vzero, accO1 = vzero;
  const int brow = l15 & 7;              // A rows m>=8: don't-care
  for (int d0 = 0; d0 < DD; d0 += 4) {
    const int dA = d0 + 2 * half;
    v2f A;                               // A[m][kk] = agg[b=m][k=w][dA+kk]
    A.x = smem[w * (DD * WAVES) + (dA + 0) * WAVES + brow];
    A.y = smem[w * (DD * WAVES) + (dA + 1) * WAVES + brow];
    const float* wp = wtk + dA * OO + l15;  // B[kk][nn] = WT[k][dA+kk][o0+nn]
    v2f B0, B1;
    B0.x = wp[0];   B0.y = wp[OO];          // o-tile 0
    B1.x = wp[16];  B1.y = wp[OO + 16];     // o-tile 1
    accO0 = __builtin_amdgcn_wmma_f32_16x16x4_f32(
        false, A, false, B0, (short)0, accO0, false, false);
    accO1 = __builtin_amdgcn_wmma_f32_16x16x4_f32(
        false, A, false, B1, (short)0, accO1, false, false);
  }

  // D-matrix vgpr j, lanes 0-15 == row b=j (rows 8-15 live in lanes 16-31, unused)
  if (lane < 16) {
    const int bq = blockIdx.x * WAVES;
    #pragma unroll
    for (int j = 0; j < WAVES; ++j) {
      float* op = out + (size_t)(bq + j) * (KK * OO) + w * OO;
      op[lane]      = accO0[j];
      op[16 + lane] = accO1[j];
    }
  }
}

extern "C" void kernel_launch(void* const* d_in, const int* in_sizes, int n_in,
                              void* d_out, int out_size, void* d_ws, size_t ws_size,
                              hipStream_t stream) {
  const float* in_feat = (const float*)d_in[0];
  const float* polar   = (const float*)d_in[1];
  const float* mL      = (const float*)d_in[2];
  const float* sdL     = (const float*)d_in[3];
  const float* mA      = (const float*)d_in[4];
  const float* W       = (const float*)d_in[5];
  float* WT  = (float*)d_ws;                 // needs 8*256*32*4 = 256 KB scratch
  float* outp = (float*)d_out;

  wt_transpose_kernel<<<(KK * OO * DD) / 256, 256, 0, stream>>>(W, WT);
  sgc_fused_kernel<<<BP / WAVES, 256, 0, stream>>>(in_feat, polar, mL, sdL, mA, WT, outp);
}